// HGCLayer_54296976556721
// MI455X (gfx1250) — compile-verified
//
#include <hip/hip_runtime.h>
#include <math.h>

// MI455X / gfx1250, wave32. All matrix math uses V_WMMA_F32_16X16X4_F32
// (exact fp32; workload is bandwidth-bound so no reason to downcast).
//
// Fragment layouts per CDNA5 ISA 7.12.2 (wave32):
//   A 16x4 f32:  lane l (lo=l&15,hi=l>>4): v[0]=A[lo][4k+2hi], v[1]=A[lo][4k+2hi+1]
//   B 4x16 f32:  lane l: v[0]=B[4k+2hi][n0+lo], v[1]=B[4k+2hi+1][n0+lo]  (assumed symmetric)
//   C/D 16x16:   lane l, vgpr r: element [M=r+8hi][N=n0+lo]

typedef __attribute__((ext_vector_type(2))) float v2f;
typedef __attribute__((ext_vector_type(8))) float v8f;

#define EPSF 1e-6f

__device__ __forceinline__ float siluf(float v) { return v / (1.f + expf(-v)); }

__device__ __forceinline__ v8f wmma4(v2f a, v2f b, v8f c) {
  // 8 args: (neg_a, A, neg_b, B, c_mod, C, reuse_a, reuse_b)
  return __builtin_amdgcn_wmma_f32_16x16x4_f32(false, a, false, b, (short)0, c,
                                               false, false);
}

// ---------------------------------------------------------------------------
// Kernel 1: h = (x @ W0^T + temb) @ W1^T    (16-node tile per wave, 4 waves/blk)
// ---------------------------------------------------------------------------
__launch_bounds__(128)
__global__ void k_gemm_h(const float* __restrict__ x, const float* __restrict__ temb,
                         const float* __restrict__ W0, const float* __restrict__ W1,
                         float* __restrict__ h, int nTiles) {
  __shared__ float W0s[4096];
  __shared__ float W1s[4096];
  __shared__ float Abuf[4][1024];
  const int tid = threadIdx.x;
  for (int i = tid; i < 4096; i += 128) { W0s[i] = W0[i]; W1s[i] = W1[i]; }
  __syncthreads();

  const int wave = tid >> 5, lane = tid & 31, lo = lane & 15, hi = lane >> 4;
  const int tile = blockIdx.x * 4 + wave;
  if (tile >= nTiles) return;
  const int m0 = tile * 16;
  float* A = Abuf[wave];

  {  // stage x tile (16x64) into per-wave LDS buffer via b128 loads
    const float4* src = (const float4*)(x + (size_t)m0 * 64);
    float4* dst = (float4*)A;
    for (int i = lane; i < 256; i += 32) dst[i] = src[i];
  }

  const v8f z8 = {0.f, 0.f, 0.f, 0.f, 0.f, 0.f, 0.f, 0.f};
  v8f acc[4] = {z8, z8, z8, z8};
  for (int k = 0; k < 16; ++k) {
    const int kb = 4 * k + 2 * hi;
    v2f a = *(const v2f*)&A[lo * 64 + kb];
#pragma unroll
    for (int nt = 0; nt < 4; ++nt) {
      v2f b = *(const v2f*)&W0s[(nt * 16 + lo) * 64 + kb];  // B[k][n] = W0[n][k]
      acc[nt] = wmma4(a, b, acc[nt]);
    }
  }
  // += temb, then write t back into the same per-wave buffer (same-wave DS is
  // in-order, no barrier needed)
#pragma unroll
  for (int nt = 0; nt < 4; ++nt) {
    const int n = nt * 16 + lo;
#pragma unroll
    for (int r = 0; r < 8; ++r)
      acc[nt][r] += temb[(size_t)(m0 + r + 8 * hi) * 64 + n];
  }
#pragma unroll
  for (int nt = 0; nt < 4; ++nt) {
    const int n = nt * 16 + lo;
#pragma unroll
    for (int r = 0; r < 8; ++r) A[(r + 8 * hi) * 64 + n] = acc[nt][r];
  }

  v8f acc2[4] = {z8, z8, z8, z8};
  for (int k = 0; k < 16; ++k) {
    const int kb = 4 * k + 2 * hi;
    v2f a = *(const v2f*)&A[lo * 64 + kb];
#pragma unroll
    for (int nt = 0; nt < 4; ++nt) {
      v2f b = *(const v2f*)&W1s[(nt * 16 + lo) * 64 + kb];
      acc2[nt] = wmma4(a, b, acc2[nt]);
    }
  }
#pragma unroll
  for (int nt = 0; nt < 4; ++nt) {
    const int n = nt * 16 + lo;
#pragma unroll
    for (int r = 0; r < 8; ++r)
      h[(size_t)(m0 + r + 8 * hi) * 64 + n] = acc2[nt][r];
  }
}

// ---------------------------------------------------------------------------
// Kernel 2: per-node  p = expmap(expmap0(proj h), transp0(., proj bias)),
//           p_tan = logmap0(p); also zero agg for this launch.
// ---------------------------------------------------------------------------
__launch_bounds__(256)
__global__ void k_node_p(const float* __restrict__ h, const float* __restrict__ bias,
                         float* __restrict__ p, float* __restrict__ pt,
                         float* __restrict__ agg, int Nn) {
  const int i = blockIdx.x * blockDim.x + threadIdx.x;
  if (i >= Nn) return;
  const float* hr = h + (size_t)i * 64;
  float u[64], pv[64];
#pragma unroll
  for (int j = 0; j < 64; j += 4) {
    float4 t = *(const float4*)&hr[j];
    u[j] = t.x; u[j + 1] = t.y; u[j + 2] = t.z; u[j + 3] = t.w;
  }
  u[0] = 0.f;  // proj_tan0
  float s2 = 0.f;
#pragma unroll
  for (int j = 1; j < 64; ++j) s2 += u[j] * u[j];
  s2 = fmaxf(s2, EPSF);
  const float n1 = sqrtf(s2);
  const float sh = sinhf(n1) / n1;
  pv[0] = coshf(n1);
#pragma unroll
  for (int j = 1; j < 64; ++j) pv[j] = sh * u[j];  // expmap0

  // v = transp0(p, proj_tan0(bias));  v0 = s,  v_j = b_j + s/(1+p0) p_j
  float s = 0.f;
#pragma unroll
  for (int j = 1; j < 64; ++j) s += pv[j] * bias[j];
  const float g = s / (1.f + pv[0]);
  u[0] = s;
#pragma unroll
  for (int j = 1; j < 64; ++j) u[j] = bias[j] + g * pv[j];

  // p = expmap(p, v)
  float lin = -u[0] * u[0];
#pragma unroll
  for (int j = 1; j < 64; ++j) lin += u[j] * u[j];
  lin = fmaxf(lin, EPSF);
  const float nv = sqrtf(lin);
  const float chv = coshf(nv), shv = sinhf(nv) / nv;
#pragma unroll
  for (int j = 0; j < 64; ++j) pv[j] = chv * pv[j] + shv * u[j];

  float* pd = p + (size_t)i * 64;
#pragma unroll
  for (int j = 0; j < 64; j += 4) {
    float4 t; t.x = pv[j]; t.y = pv[j + 1]; t.z = pv[j + 2]; t.w = pv[j + 3];
    *(float4*)&pd[j] = t;
  }
  // p_tan = logmap0(p)
  const float d = acoshf(fmaxf(pv[0], 1.f + EPSF));
  float n2 = 0.f;
#pragma unroll
  for (int j = 1; j < 64; ++j) n2 += pv[j] * pv[j];
  n2 = fmaxf(n2, EPSF);
  const float c = d / sqrtf(n2);
  float* td = pt + (size_t)i * 64;
  float* ad = agg + (size_t)i * 64;
  const float4 z4 = {0.f, 0.f, 0.f, 0.f};
#pragma unroll
  for (int j = 0; j < 64; j += 4) {
    float4 t;
    t.x = (j == 0) ? 0.f : c * pv[j];
    t.y = c * pv[j + 1]; t.z = c * pv[j + 2]; t.w = c * pv[j + 3];
    *(float4*)&td[j] = t;
    *(float4*)&ad[j] = z4;  // zero agg every launch (no cross-call state)
  }
}

// ---------------------------------------------------------------------------
// Kernel 3: per-edge a = clip(-linner(p[row], p[col]), 1+eps)  (2 lanes/edge)
// ---------------------------------------------------------------------------
__launch_bounds__(256)
__global__ void k_edge_geo(const float* __restrict__ p, const int* __restrict__ row,
                           const int* __restrict__ col, float* __restrict__ aE, int Ne) {
  const int t = blockIdx.x * blockDim.x + threadIdx.x;
  const int e = t >> 1, half = t & 1;
  if (e >= Ne) return;
  const float* pr = p + (size_t)row[e] * 64 + half * 32;
  const float* pc = p + (size_t)col[e] * 64 + half * 32;
  float dot = 0.f;
#pragma unroll
  for (int j = 0; j < 32; j += 4) {
    float4 a4 = *(const float4*)&pr[j];
    float4 b4 = *(const float4*)&pc[j];
    dot += a4.x * b4.x + a4.y * b4.y + a4.z * b4.z + a4.w * b4.w;
  }
  if (half == 0) dot -= 2.f * pr[0] * pc[0];  // Lorentz sign on component 0
  dot += __shfl_xor(dot, 1);
  if (half == 0) aE[e] = fmaxf(-dot, 1.f + EPSF);
}

// ---------------------------------------------------------------------------
// Kernel 4: attention.  a@Wa1^T split: pt[row]@WaA^T + pt[col]@WaB^T + geo*wc
// then silu, dot with Wa2 (shuffle-reduce), sigmoid * edge_mask -> att[e]
// 16-edge tile per wave, 2 waves/block.
// ---------------------------------------------------------------------------
__launch_bounds__(64)
__global__ void k_edge_att(const float* __restrict__ pt, const int* __restrict__ row,
                           const int* __restrict__ col, const float* __restrict__ aE,
                           const float* __restrict__ Wa1, const float* __restrict__ ba1,
                           const float* __restrict__ Wa2, const float* __restrict__ ba2,
                           const float* __restrict__ emask, float* __restrict__ att,
                           int nTiles) {
  __shared__ float WaA[4096], WaB[4096], wc[64], b1[64], w2[64];
  __shared__ float A1[2][1024], A2[2][1024], gs[2][16];
  const int tid = threadIdx.x;
  for (int i = tid; i < 4096; i += 64) {
    const int n = i >> 6, k = i & 63;
    WaA[i] = Wa1[n * 129 + k];
    WaB[i] = Wa1[n * 129 + 64 + k];
  }
  if (tid < 64) { wc[tid] = Wa1[tid * 129 + 128]; b1[tid] = ba1[tid]; w2[tid] = Wa2[tid]; }
  __syncthreads();

  const int wave = tid >> 5, lane = tid & 31, lo = lane & 15, hi = lane >> 4;
  const int tile = blockIdx.x * 2 + wave;
  if (tile >= nTiles) return;
  const int e0 = tile * 16;
  float* a1 = A1[wave];
  float* a2 = A2[wave];
  {  // gather p_tan rows, 2 lanes per edge
    const int m = lane >> 1, half = lane & 1, e = e0 + m;
    const float* s1 = pt + (size_t)row[e] * 64 + half * 32;
    const float* s2 = pt + (size_t)col[e] * 64 + half * 32;
#pragma unroll
    for (int j = 0; j < 32; j += 4) {
      *(float4*)&a1[m * 64 + half * 32 + j] = *(const float4*)&s1[j];
      *(float4*)&a2[m * 64 + half * 32 + j] = *(const float4*)&s2[j];
    }
    if (half == 0) gs[wave][m] = acoshf(aE[e]);  // geo = dist
  }

  const v8f z8 = {0.f, 0.f, 0.f, 0.f, 0.f, 0.f, 0.f, 0.f};
  v8f acc[4] = {z8, z8, z8, z8};
  for (int k = 0; k < 16; ++k) {
    const int kb = 4 * k + 2 * hi;
    v2f va = *(const v2f*)&a1[lo * 64 + kb];
    v2f vb = *(const v2f*)&a2[lo * 64 + kb];
#pragma unroll
    for (int nt = 0; nt < 4; ++nt) {
      v2f wA = *(const v2f*)&WaA[(nt * 16 + lo) * 64 + kb];
      acc[nt] = wmma4(va, wA, acc[nt]);
      v2f wB = *(const v2f*)&WaB[(nt * 16 + lo) * 64 + kb];
      acc[nt] = wmma4(vb, wB, acc[nt]);
    }
  }

  float part[8];
#pragma unroll
  for (int r = 0; r < 8; ++r) part[r] = 0.f;
#pragma unroll
  for (int nt = 0; nt < 4; ++nt) {
    const int n = nt * 16 + lo;
    const float bn = b1[n], wcn = wc[n], w2n = w2[n];
#pragma unroll
    for (int r = 0; r < 8; ++r) {
      float v = acc[nt][r] + gs[wave][r + 8 * hi] * wcn + bn;
      part[r] += siluf(v) * w2n;
    }
  }
  const float bb2 = ba2[0];
#pragma unroll
  for (int r = 0; r < 8; ++r) {
    float v = part[r];
    v += __shfl_xor(v, 1);
    v += __shfl_xor(v, 2);
    v += __shfl_xor(v, 4);
    v += __shfl_xor(v, 8);  // stays within each 16-lane half (wave32)
    if (lo == 0) {
      const int e = e0 + r + 8 * hi;
      att[e] = (1.f / (1.f + expf(-(v + bb2)))) * emask[e];
    }
  }
}

// ---------------------------------------------------------------------------
// Kernel 5: messages.  lt = transp0back(pr, logmap(pr,pc)); two WMMA GEMMs +
// silu; scale by att; atomic scatter-add into agg (L2-resident, 12.8 MB).
// 16-edge tile per wave, 4 waves/block.
// ---------------------------------------------------------------------------
__launch_bounds__(128)
__global__ void k_edge_msg(const float* __restrict__ p, const int* __restrict__ row,
                           const int* __restrict__ col, const float* __restrict__ aE,
                           const float* __restrict__ att, const float* __restrict__ We1,
                           const float* __restrict__ be1, const float* __restrict__ We2,
                           const float* __restrict__ be2, float* __restrict__ agg,
                           int nTiles) {
  __shared__ float W1s[4096], W2s[4096], b1[64], b2[64];
  __shared__ float buf[4][1024], ats[4][16];
  __shared__ int ridx[4][16];
  const int tid = threadIdx.x;
  for (int i = tid; i < 4096; i += 128) { W1s[i] = We1[i]; W2s[i] = We2[i]; }
  if (tid < 64) { b1[tid] = be1[tid]; b2[tid] = be2[tid]; }
  __syncthreads();

  const int wave = tid >> 5, lane = tid & 31, lo = lane & 15, hi = lane >> 4;
  const int tile = blockIdx.x * 4 + wave;
  if (tile >= nTiles) return;
  const int e0 = tile * 16;
  float* B = buf[wave];

  {  // lt into per-wave LDS; elementwise given a=aE[e]
    const int m = lane >> 1, half = lane & 1, e = e0 + m;
    const int ri = row[e], ci = col[e];
    if (half == 0) { ridx[wave][m] = ri; ats[wave][m] = att[e]; }
    const float* pr = p + (size_t)ri * 64;
    const float* pc = p + (size_t)ci * 64;
    const float a = aE[e];
    const float pr0 = pr[0], pc0 = pc[0];
    const float co = acoshf(a) * rsqrtf(a * a - 1.f);  // a >= 1+eps
    const float v0 = co * (pc0 - a * pr0);
    const float g = v0 / (1.f + pr0);
    const float* prh = pr + half * 32;
    const float* pch = pc + half * 32;
#pragma unroll
    for (int j = 0; j < 32; j += 4) {
      float4 x4 = *(const float4*)&prh[j];
      float4 y4 = *(const float4*)&pch[j];
      float4 o;
      o.x = co * (y4.x - a * x4.x) - g * x4.x;
      o.y = co * (y4.y - a * x4.y) - g * x4.y;
      o.z = co * (y4.z - a * x4.z) - g * x4.z;
      o.w = co * (y4.w - a * x4.w) - g * x4.w;
      *(float4*)&B[m * 64 + half * 32 + j] = o;
    }
    if (half == 0) B[m * 64] = 0.f;  // lt_0 is exactly 0
  }

  const v8f z8 = {0.f, 0.f, 0.f, 0.f, 0.f, 0.f, 0.f, 0.f};
  v8f acc[4] = {z8, z8, z8, z8};
  for (int k = 0; k < 16; ++k) {
    const int kb = 4 * k + 2 * hi;
    v2f a = *(const v2f*)&B[lo * 64 + kb];
#pragma unroll
    for (int nt = 0; nt < 4; ++nt) {
      v2f b = *(const v2f*)&W1s[(nt * 16 + lo) * 64 + kb];
      acc[nt] = wmma4(a, b, acc[nt]);
    }
  }
  // t1 = silu(. + be1) written back to same buffer (same-wave DS in-order)
#pragma unroll
  for (int nt = 0; nt < 4; ++nt) {
    const int n = nt * 16 + lo;
#pragma unroll
    for (int r = 0; r < 8; ++r)
      B[(r + 8 * hi) * 64 + n] = siluf(acc[nt][r] + b1[n]);
  }
  v8f acc2[4] = {z8, z8, z8, z8};
  for (int k = 0; k < 16; ++k) {
    const int kb = 4 * k + 2 * hi;
    v2f a = *(const v2f*)&B[lo * 64 + kb];
#pragma unroll
    for (int nt = 0; nt < 4; ++nt) {
      v2f b = *(const v2f*)&W2s[(nt * 16 + lo) * 64 + kb];
      acc2[nt] = wmma4(a, b, acc2[nt]);
    }
  }
#pragma unroll
  for (int nt = 0; nt < 4; ++nt) {
    const int n = nt * 16 + lo;
    const float bn = b2[n];
#pragma unroll
    for (int r = 0; r < 8; ++r) {
      const int m = r + 8 * hi;
      const float val = (acc2[nt][r] + bn) * ats[wave][m];
      atomicAdd(&agg[(size_t)ridx[wave][m] * 64 + n], val);
    }
  }
}

// ---------------------------------------------------------------------------
// Kernel 6: per-node  p2 = expmap(p, transp0(p, proj agg)); out = layernorm/
// silu of logmap0(p2)
// ---------------------------------------------------------------------------
__launch_bounds__(256)
__global__ void k_node_out(const float* __restrict__ p, const float* __restrict__ agg,
                           const float* __restrict__ ln_g, const float* __restrict__ ln_b,
                           float* __restrict__ out, int Nn) {
  const int i = blockIdx.x * blockDim.x + threadIdx.x;
  if (i >= Nn) return;
  const float* pr = p + (size_t)i * 64;
  const float* ar = agg + (size_t)i * 64;
  float pv[64], u[64];
#pragma unroll
  for (int j = 0; j < 64; j += 4) {
    float4 t = *(const float4*)&pr[j];
    pv[j] = t.x; pv[j + 1] = t.y; pv[j + 2] = t.z; pv[j + 3] = t.w;
    float4 s = *(const float4*)&ar[j];
    u[j] = s.x; u[j + 1] = s.y; u[j + 2] = s.z; u[j + 3] = s.w;
  }
  u[0] = 0.f;  // proj_tan0(agg)
  float s = 0.f;
#pragma unroll
  for (int j = 1; j < 64; ++j) s += pv[j] * u[j];
  const float g = s / (1.f + pv[0]);
  u[0] = s;
#pragma unroll
  for (int j = 1; j < 64; ++j) u[j] = u[j] + g * pv[j];  // transp0
  float lin = -u[0] * u[0];
#pragma unroll
  for (int j = 1; j < 64; ++j) lin += u[j] * u[j];
  lin = fmaxf(lin, EPSF);
  const float nv = sqrtf(lin);
  const float chv = coshf(nv), shv = sinhf(nv) / nv;
#pragma unroll
  for (int j = 0; j < 64; ++j) pv[j] = chv * pv[j] + shv * u[j];  // expmap
  // logmap0
  const float d = acoshf(fmaxf(pv[0], 1.f + EPSF));
  float n2 = 0.f;
#pragma unroll
  for (int j = 1; j < 64; ++j) n2 += pv[j] * pv[j];
  n2 = fmaxf(n2, EPSF);
  const float c = d / sqrtf(n2);
#pragma unroll
  for (int j = 1; j < 64; ++j) u[j] = c * pv[j];  // out spatial part
  // layernorm over 63 spatial dims
  float mu = 0.f;
#pragma unroll
  for (int j = 1; j < 64; ++j) mu += u[j];
  mu *= (1.f / 63.f);
  float var = 0.f;
#pragma unroll
  for (int j = 1; j < 64; ++j) { const float dlt = u[j] - mu; var += dlt * dlt; }
  var *= (1.f / 63.f);
  const float rs = rsqrtf(var + 1e-5f);
  float* od = out + (size_t)i * 64;
  od[0] = 0.f;  // silu(0)=0, then proj_tan0 sets 0
#pragma unroll
  for (int j = 1; j < 64; ++j)
    od[j] = siluf((u[j] - mu) * rs * ln_g[j - 1] + ln_b[j - 1]);
}

// ---------------------------------------------------------------------------
extern "C" void kernel_launch(void* const* d_in, const int* in_sizes, int n_in,
                              void* d_out, int out_size, void* d_ws, size_t ws_size,
                              hipStream_t stream) {
  const float* x     = (const float*)d_in[0];
  const int*   edges = (const int*)d_in[1];
  // d_in[2] = node_mask (unused by the reference math)
  const float* emask = (const float*)d_in[3];
  const float* temb  = (const float*)d_in[4];
  const float* W0    = (const float*)d_in[5];
  const float* W1    = (const float*)d_in[6];
  const float* bias  = (const float*)d_in[7];
  const float* We1   = (const float*)d_in[8];
  const float* be1   = (const float*)d_in[9];
  const float* We2   = (const float*)d_in[10];
  const float* be2   = (const float*)d_in[11];
  const float* Wa1   = (const float*)d_in[12];
  const float* ba1   = (const float*)d_in[13];
  const float* Wa2   = (const float*)d_in[14];
  const float* ba2   = (const float*)d_in[15];
  const float* ln_g  = (const float*)d_in[16];
  const float* ln_b  = (const float*)d_in[17];

  const int N = in_sizes[0] / 64;
  const int E = in_sizes[1] / 2;
  const int* row = edges;
  const int* col = edges + E;

  float* ws = (float*)d_ws;
  const size_t N64 = (size_t)N * 64;
  float* h    = ws;
  float* p    = ws + N64;
  float* pt   = ws + 2 * N64;
  float* agg  = ws + 3 * N64;
  float* aE   = ws + 4 * N64;
  float* attb = aE + E;   // total ws: 4*N*64 + 2*E floats (~58 MB)
  float* out  = (float*)d_out;

  const int ntN = (N + 15) / 16;   // N=50000 -> 3125 tiles (exact)
  const int ntE = (E + 15) / 16;   // E=800000 -> 50000 tiles (exact)

  k_gemm_h <<<(ntN + 3) / 4, 128, 0, stream>>>(x, temb, W0, W1, h, ntN);
  k_node_p <<<(N + 255) / 256, 256, 0, stream>>>(h, bias, p, pt, agg, N);
  k_edge_geo<<<(2 * E + 255) / 256, 256, 0, stream>>>(p, row, col, aE, E);
  k_edge_att<<<(ntE + 1) / 2, 64, 0, stream>>>(pt, row, col, aE, Wa1, ba1, Wa2,
                                               ba2, emask, attb, ntE);
  k_edge_msg<<<(ntE + 3) / 4, 128, 0, stream>>>(p, row, col, aE, attb, We1, be1,
                                                We2, be2, agg, ntE);
  k_node_out<<<(N + 255) / 256, 256, 0, stream>>>(p, agg, ln_g, ln_b, out, N);
}